// Correlation_8821862826318
// MI455X (gfx1250) — compile-verified
//
#include <hip/hip_runtime.h>

typedef __attribute__((ext_vector_type(2))) float v2f;
typedef __attribute__((ext_vector_type(4))) float v4f;
typedef __attribute__((ext_vector_type(8))) float v8f;

#define B_  8
#define C_  32
#define H_  256
#define W_  512
#define D_  48
#define CS  (H_ * W_)      // channel stride in elements
#define PITCH 65           // LDS row pitch (floats) -> conflict-free diagonal scatter

// One block = 4 waves = 64 output columns (tg), one (b,h) row.
// Each wave: 16-col strip ti, 4 band tiles (Delta=0..3), K=32 via 8x wmma_f32_16x16x4_f32.
__global__ __launch_bounds__(128) void corr_wmma_kernel(const float* __restrict__ x,
                                                        const float* __restrict__ y,
                                                        float* __restrict__ out) {
    __shared__ float lds[D_ * PITCH];   // 48 * 65 * 4 = 12480 B

    const int t    = threadIdx.x;
    const int lane = t & 31;
    const int l16  = lane & 15;
    const int half = lane >> 4;

    const int blk = blockIdx.x;
    const int tg  = blk & 7;           // 8 groups of 64 columns
    const int h   = (blk >> 3) & (H_ - 1);
    const int b   = blk >> 11;

    // ---- phase 0: zero cost slab (covers w<d region and absent tj<0 tiles)
    for (int i = t; i < D_ * PITCH; i += 128) lds[i] = 0.0f;
    __syncthreads();

    // wave-uniform strip index (readfirstlane -> scalar branches, EXEC untouched)
    const int ti = __builtin_amdgcn_readfirstlane(t >> 5) + tg * 4;  // 0..31
    const int w0 = ti << 4;

    const int plane = (b * C_ * H_ + h) * W_;     // fits in 32-bit
    const float* xp = x + plane + w0 + l16;       // A: M = l16, K pair selected by half
    const float* yp = y + plane + w0 + l16;       // B: N = l16, K pair selected by half

    // clamped B pointers: invalid Delta tiles read in-bounds garbage, never stored
    const float* yq[4];
    yq[0] = yp;
    yq[1] = (ti >= 1) ? (yp - 16) : yp;
    yq[2] = (ti >= 2) ? (yp - 32) : yp;
    yq[3] = (ti >= 3) ? (yp - 48) : yp;

    v8f acc[4] = {};
    const int ch = half * 2;   // lanes 0-15: K=0,1 ; lanes 16-31: K=2,3

    #pragma unroll
    for (int k = 0; k < 8; ++k) {
        const int co = (4 * k + ch) * CS;
        v2f a;
        a.x = xp[co];
        a.y = xp[co + CS];
        #pragma unroll
        for (int dlt = 0; dlt < 4; ++dlt) {
            v2f bb;
            bb.x = yq[dlt][co];
            bb.y = yq[dlt][co + CS];
            acc[dlt] = __builtin_amdgcn_wmma_f32_16x16x4_f32(
                false, a, false, bb, (short)0, acc[dlt], false, false);
        }
    }

    // ---- phase 1: scatter band diagonals into LDS cost[d][w_local]
    // element (m,n) of tile Delta: d = 16*Delta + m - n, w_local = 16*waveId + m
    const int wl_base = (t >> 5) << 4;
    #pragma unroll
    for (int dlt = 0; dlt < 4; ++dlt) {
        if (ti >= dlt) {                       // uniform per wave
            #pragma unroll
            for (int v = 0; v < 8; ++v) {
                const int m = v + 8 * half;
                const int d = 16 * dlt + m - l16;
                if (d >= 0 && d < D_) {
                    lds[d * PITCH + wl_base + m] = acc[dlt][v];
                }
            }
        }
    }
    __syncthreads();

    // ---- phase 2: coalesced store, 48 rows x 64 floats, apply mean (1/32)
    float* op = out + (b * D_ * H_ + h) * W_ + tg * 64;
    #pragma unroll
    for (int i = 0; i < 6; ++i) {
        const int idx = t + i * 128;       // 0..767
        const int row = idx >> 4;          // d
        const int q   = (idx & 15) << 2;   // float offset within row
        const int base = row * PITCH + q;
        v4f val;
        val.x = lds[base + 0] * 0.03125f;
        val.y = lds[base + 1] * 0.03125f;
        val.z = lds[base + 2] * 0.03125f;
        val.w = lds[base + 3] * 0.03125f;
        *(v4f*)(op + row * (H_ * W_) + q) = val;
    }
}

extern "C" void kernel_launch(void* const* d_in, const int* in_sizes, int n_in,
                              void* d_out, int out_size, void* d_ws, size_t ws_size,
                              hipStream_t stream) {
    const float* x = (const float*)d_in[0];
    const float* y = (const float*)d_in[1];
    float* out = (float*)d_out;
    (void)in_sizes; (void)n_in; (void)out_size; (void)d_ws; (void)ws_size;

    dim3 grid(B_ * H_ * (W_ / 64));   // 8 * 256 * 8 = 16384 blocks
    dim3 block(128);                  // 4 waves
    corr_wmma_kernel<<<grid, block, 0, stream>>>(x, y, out);
}